// DetectionLoss_85693187489837
// MI455X (gfx1250) — compile-verified
//
#include <hip/hip_runtime.h>
#include <math.h>

typedef __attribute__((ext_vector_type(2))) float v2f;
typedef __attribute__((ext_vector_type(8))) float v8f;

#define NC 80
#define NGT 128
#define DIST_THRESH 2.5f

// Branch-free xor-swap of a 64-bit value across lanes via DS_SWIZZLE_B32
// (group-of-32 mode: offset[15]=0, xor_mask=bits[14:10], and_mask=0x1f).
template <int MASK>
__device__ __forceinline__ unsigned long long swz64(unsigned long long k)
{
    constexpr int pat = (MASK << 10) | 0x1f;
    unsigned lo = (unsigned)__builtin_amdgcn_ds_swizzle((int)(unsigned)(k & 0xffffffffull), pat);
    unsigned hi = (unsigned)__builtin_amdgcn_ds_swizzle((int)(unsigned)(k >> 32), pat);
    return ((unsigned long long)hi << 32) | lo;
}

// One block per (batch, level). 256 threads = 8 waves; wave w owns gts [16w,16w+16).
// Phase A: V_WMMA_F32_16X16X4_F32 computes 16x16 tiles of d2 = |g|^2 - 2 g.p + |p|^2
//          A[m,:] = [-2gx, -2gy, |g|^2, 1]  (f32 A 16x4: lanes<16 hold K0/K1, lanes>=16 K2/K3)
//          B[:,n] = [ px,   py,   1, |p|^2] (f32 B 4x16: lanes<16 rows K0/K1, lanes>=16 K2/K3)
// Phase B: gather matched rows (b128 reads), log-softmax CE + L1, block reduce -> d_ws.
__global__ __launch_bounds__(256)
void det_loss_level_kernel(const float* __restrict__ cls0, const float* __restrict__ cls1,
                           const float* __restrict__ cls2,
                           const float* __restrict__ reg0, const float* __restrict__ reg1,
                           const float* __restrict__ reg2,
                           const float* __restrict__ gt_boxes, const int* __restrict__ gt_labels,
                           float* __restrict__ partials)
{
    const int b   = blockIdx.x;   // batch
    const int lvl = blockIdx.y;   // pyramid level
    const float* cls; const float* reg; int HW;
    if (lvl == 0)      { cls = cls0; reg = reg0; HW = 25600; }
    else if (lvl == 1) { cls = cls1; reg = reg1; HW = 6400;  }
    else               { cls = cls2; reg = reg2; HW = 1600;  }

    __shared__ float sD[NGT];
    __shared__ int   sI[NGT];
    __shared__ float red0[256], red1[256], red2[256];

    const int  tid  = threadIdx.x;
    const int  wave = tid >> 5;
    const int  lane = tid & 31;
    const int  col  = lane & 15;
    const bool hi   = lane >= 16;

    // ---------------- Phase A: WMMA pairwise d2 + per-gt argmin ----------------
    const int    gt = wave * 16 + col;
    const float2 g  = *(const float2*)(gt_boxes + ((size_t)b * NGT + gt) * 4);

    v2f A;                                                 // 16x4 f32 A-matrix operand
    A.x = hi ? (g.x * g.x + g.y * g.y) : (-2.0f * g.x);    // K=2 / K=0
    A.y = hi ? 1.0f                    : (-2.0f * g.y);    // K=3 / K=1

    float bestD[8];
    int   bestI[8];
#pragma unroll
    for (int r = 0; r < 8; ++r) { bestD[r] = 3.4e38f; bestI[r] = 0; }

    const float* regB   = reg + (size_t)b * HW * 4;
    const int    ntiles = HW >> 4;                         // HW % 16 == 0 at every level

    // Software pipeline: prefetch tile t+1 (scalar-clamped, no divergence) while
    // the WMMA + argmin chain consumes tile t.
    float2 p = *(const float2*)(regB + (size_t)col * 4);   // tile 0
#pragma unroll 2
    for (int t = 0; t < ntiles; ++t) {
        const int    tn = (t + 1 < ntiles) ? (t + 1) : t;
        const float2 pn = *(const float2*)(regB + ((size_t)(tn << 4) + col) * 4);

        v2f Bm;                                            // 4x16 f32 B-matrix operand
        Bm.x = hi ? 1.0f                    : p.x;         // K=2 / K=0
        Bm.y = hi ? (p.x * p.x + p.y * p.y) : p.y;         // K=3 / K=1
        v8f c = {};
        v8f d = __builtin_amdgcn_wmma_f32_16x16x4_f32(
            /*neg_a=*/false, A, /*neg_b=*/false, Bm,
            /*c_mod=*/(short)0, c, /*reuse_a=*/false, /*reuse_b=*/false);

        const int n = (t << 4) + col;
#pragma unroll
        for (int r = 0; r < 8; ++r) {
            if (d[r] < bestD[r]) { bestD[r] = d[r]; bestI[r] = n; }
        }
        p = pn;
    }

    // Cross-lane min over the 16 anchor columns of each half-wave.
    // Pack (d2_bits << 32) | idx: d2 >= 0 so float bits are order-preserving and
    // unsigned u64 '<' gives jnp.argmin's first-occurrence tie-break for free.
#pragma unroll
    for (int r = 0; r < 8; ++r) {
        const float dc = fmaxf(bestD[r], 0.0f);
        unsigned long long k =
            ((unsigned long long)__float_as_uint(dc) << 32) | (unsigned)bestI[r];
        unsigned long long o;
        o = swz64<1>(k); k = (o < k) ? o : k;
        o = swz64<2>(k); k = (o < k) ? o : k;
        o = swz64<4>(k); k = (o < k) ? o : k;
        o = swz64<8>(k); k = (o < k) ? o : k;
        bestD[r] = __uint_as_float((unsigned)(k >> 32));
        bestI[r] = (int)(unsigned)(k & 0xffffffffull);
    }

    // lane 0 holds final rows m=0..7 of this wave's gt tile, lane 16 rows m=8..15.
    if (lane == 0 || lane == 16) {
#pragma unroll
        for (int r = 0; r < 8; ++r) {
            const int m = r + (hi ? 8 : 0);
            sD[wave * 16 + m] = bestD[r];
            sI[wave * 16 + m] = bestI[r];
        }
    }
    __syncthreads();

    // ---------------- Phase B: gather + CE + L1 (b128 row reads) ----------------
    float ce_w = 0.0f, l1_w = 0.0f, w = 0.0f;
    if (tid < NGT) {
        const int   row = sI[tid];
        const float md  = sD[tid];
        w = (sqrtf(md) < DIST_THRESH) ? 1.0f : 0.0f;

        // class row: 80 floats = 20 float4 (rows are 320B, 16B-aligned)
        const float4* c4 = (const float4*)(cls + ((size_t)b * HW + row) * NC);
        float mx = -3.4e38f;
#pragma unroll 4
        for (int k = 0; k < NC / 4; ++k) {
            const float4 v = c4[k];
            mx = fmaxf(mx, fmaxf(fmaxf(v.x, v.y), fmaxf(v.z, v.w)));
        }
        float se = 0.0f;
#pragma unroll 4
        for (int k = 0; k < NC / 4; ++k) {
            const float4 v = c4[k];
            se += __expf(v.x - mx) + __expf(v.y - mx) + __expf(v.z - mx) + __expf(v.w - mx);
        }
        const int   label = gt_labels[b * NGT + tid];
        const float logit = ((const float*)c4)[label];
        const float ce    = __logf(se) + mx - logit;

        const float4 rv = *(const float4*)(regB + (size_t)row * 4);
        const float4 gv = *(const float4*)(gt_boxes + ((size_t)b * NGT + tid) * 4);
        const float  l1 = fabsf(rv.x - gv.x) + fabsf(rv.y - gv.y) +
                          fabsf(rv.z - gv.z) + fabsf(rv.w - gv.w);

        ce_w = ce * w;
        l1_w = l1 * w;
    }
    red0[tid] = ce_w; red1[tid] = l1_w; red2[tid] = w;
    __syncthreads();
    for (int s = 128; s > 0; s >>= 1) {
        if (tid < s) {
            red0[tid] += red0[tid + s];
            red1[tid] += red1[tid + s];
            red2[tid] += red2[tid + s];
        }
        __syncthreads();
    }
    if (tid == 0) {
        const int blk = blockIdx.y * gridDim.x + blockIdx.x;
        partials[blk * 3 + 0] = red0[0];
        partials[blk * 3 + 1] = red1[0];
        partials[blk * 3 + 2] = red2[0];
    }
}

__global__ __launch_bounds__(64)
void det_loss_finalize(const float* __restrict__ partials, float* __restrict__ out, int nblk)
{
    __shared__ float s0[64], s1[64], s2[64];
    const int tid = threadIdx.x;
    float a = 0.0f, bb = 0.0f, c = 0.0f;
    for (int i = tid; i < nblk; i += 64) {
        a  += partials[i * 3 + 0];
        bb += partials[i * 3 + 1];
        c  += partials[i * 3 + 2];
    }
    s0[tid] = a; s1[tid] = bb; s2[tid] = c;
    __syncthreads();
    for (int s = 32; s > 0; s >>= 1) {
        if (tid < s) { s0[tid] += s0[tid + s]; s1[tid] += s1[tid + s]; s2[tid] += s2[tid + s]; }
        __syncthreads();
    }
    if (tid == 0) {
        const float np    = s2[0];
        const float denom = fmaxf(np, 1.0f);
        out[0] = s0[0] / denom;
        out[1] = s1[0] / denom;
        out[2] = np;
    }
}

extern "C" void kernel_launch(void* const* d_in, const int* in_sizes, int n_in,
                              void* d_out, int out_size, void* d_ws, size_t ws_size,
                              hipStream_t stream)
{
    const float* cls0 = (const float*)d_in[0];
    const float* cls1 = (const float*)d_in[1];
    const float* cls2 = (const float*)d_in[2];
    const float* reg0 = (const float*)d_in[3];
    const float* reg1 = (const float*)d_in[4];
    const float* reg2 = (const float*)d_in[5];
    const float* gtb  = (const float*)d_in[6];
    const int*   gtl  = (const int*)d_in[7];
    float* partials   = (float*)d_ws;   // 48 blocks * 3 floats = 576 B

    dim3 grid(16, 3);  // (batch, level)
    det_loss_level_kernel<<<grid, 256, 0, stream>>>(cls0, cls1, cls2,
                                                    reg0, reg1, reg2,
                                                    gtb, gtl, partials);
    det_loss_finalize<<<1, 64, 0, stream>>>(partials, (float*)d_out, 48);
}